// CriticRNN_89412629168157
// MI455X (gfx1250) — compile-verified
//
#include <hip/hip_runtime.h>
#include <hip/hip_bf16.h>

typedef float v2f __attribute__((ext_vector_type(2)));
typedef float v8f __attribute__((ext_vector_type(8)));

#define T_DIM 2048
#define B_DIM 256
#define D_DIM 256
#define H_DIM 8
#define G_DIM 32   // 4*H

// ---------------------------------------------------------------------------
// Phase 1: x_gates[T*B, 32] = x[T*B, 256] @ W_ih^T[256, 32] + b_ih
// f32 WMMA 16x16x4. Block = 256 threads (8 waves); wave -> one 16x32 C tile.
// W_ih^T staged in LDS as K-pairs: Bp[p][n] = { W_ih[n][2p], W_ih[n][2p+1] }.
// ---------------------------------------------------------------------------
__global__ void __launch_bounds__(256) xgates_gemm_kernel(
    const float* __restrict__ x,      // [T*B, 256]
    const float* __restrict__ Wih,    // [32, 256] row-major
    const float* __restrict__ bih,    // [32]
    float* __restrict__ xg)           // [T*B, 32]
{
    __shared__ float Bp[128][32][2];  // 32 KB

    const int tid = threadIdx.x;

    // Stage W_ih (32x256 f32 = 8192 floats) into LDS, paired along K.
    #pragma unroll
    for (int it = 0; it < 8; ++it) {
        int chunk = tid + it * 256;        // 0..2047 b128 chunks
        int n  = chunk >> 6;               // 0..31 (gate column)
        int k4 = (chunk & 63) << 2;        // 0..252 step 4
        const float4 v = *(const float4*)(Wih + (long)n * D_DIM + k4);
        int p = k4 >> 1;                   // even pair index
        Bp[p][n][0]     = v.x;  Bp[p][n][1]     = v.y;
        Bp[p + 1][n][0] = v.z;  Bp[p + 1][n][1] = v.w;
    }
    __syncthreads();

    const int wave = tid >> 5;
    const int lane = tid & 31;
    const int l16  = lane & 15;            // M row within tile (A), N col (B/C)
    const int hh   = lane >> 4;            // K-half selector

    const long m0 = (long)blockIdx.x * 128 + (long)wave * 16;
    const float* arow = x + (m0 + l16) * (long)D_DIM + 2 * hh;

    v8f acc0 = {};
    v8f acc1 = {};

    // K loop: 64 iterations of K=4, two N tiles per iteration.
    #pragma unroll 4
    for (int k = 0; k < D_DIM; k += 4) {
        // A 16x4: lane<16 holds {A[m][k],A[m][k+1]}, lane>=16 {A[m][k+2],A[m][k+3]}
        v2f a  = *(const v2f*)(arow + k);
        // B 4x16: lane<16 holds {B[k][n],B[k+1][n]}, lane>=16 {B[k+2][n],B[k+3][n]}
        v2f b0 = *(const v2f*)&Bp[(k >> 1) + hh][l16][0];
        v2f b1 = *(const v2f*)&Bp[(k >> 1) + hh][l16 + 16][0];
        acc0 = __builtin_amdgcn_wmma_f32_16x16x4_f32(false, a, false, b0,
                                                     (short)0, acc0, false, false);
        acc1 = __builtin_amdgcn_wmma_f32_16x16x4_f32(false, a, false, b1,
                                                     (short)0, acc1, false, false);
    }

    // Epilogue: C layout — VGPR r, lane L: M = r + 8*(L/16), N = L%16 (+16 tile1)
    const float bias0 = bih[l16];
    const float bias1 = bih[l16 + 16];
    float* outp = xg + (m0 + 8 * hh) * (long)G_DIM;
    #pragma unroll
    for (int r = 0; r < 8; ++r) {
        outp[(long)r * G_DIM + l16]      = acc0[r] + bias0;
        outp[(long)r * G_DIM + l16 + 16] = acc1[r] + bias1;
    }
}

// ---------------------------------------------------------------------------
// Phase 2: sequential LSTM scan + fused output projection.
// Register-resident, zero LDS, zero barriers. 8 lanes per batch element:
// lane = (b_local[2b], h_idx[3b]); each lane owns c,h for one hidden unit and
// the 4 gate rows W_hh[h_idx + 8q][:] (32 regs). h-broadcast via __shfl within
// the wave; projection via __shfl_xor butterfly. Loads prefetched 1 step ahead
// (x_gates is 64 MB -> resident in the 192 MB L2).
// ---------------------------------------------------------------------------
__device__ __forceinline__ float fast_sigmoid(float x) {
    return 1.0f / (1.0f + __expf(-x));
}
__device__ __forceinline__ float fast_tanh(float x) {
    // robust at |x| large: exp->inf gives exactly +/-1, no NaN
    return 1.0f - 2.0f / (1.0f + __expf(2.0f * x));
}

__global__ void __launch_bounds__(256) lstm_scan_kernel(
    const float* __restrict__ xg,     // [T, B, 32] (bias already fused)
    const int*   __restrict__ reset,  // [T, B]
    const float* __restrict__ Whh,    // [32, 8]
    const float* __restrict__ bhh,    // [32]
    const float* __restrict__ Wproj,  // [1, 8]
    const float* __restrict__ bproj,  // [1]
    float* __restrict__ out)          // [T, B]
{
    const int tid  = threadIdx.x;
    const int wave = tid >> 5;
    const int lane = tid & 31;
    const int hid  = lane & 7;         // hidden index 0..7
    const int lb   = lane & 24;        // base lane of this batch's 8-lane group
    const int b    = blockIdx.x * 32 + wave * 4 + (lane >> 3);

    // Per-lane weights: w[q][j] = W_hh[hid + 8q][j]  (gate order i,f,g,o)
    float w[4][8], bias[4];
    #pragma unroll
    for (int q = 0; q < 4; ++q) {
        bias[q] = bhh[hid + 8 * q];
        #pragma unroll
        for (int j = 0; j < 8; ++j)
            w[q][j] = Whh[(hid + 8 * q) * H_DIM + j];
    }
    const float wp = Wproj[hid];
    const float bp = bproj[0];

    float h = 0.0f, c = 0.0f;

    // Prefetch t = 0
    float nxv[4];
    #pragma unroll
    for (int q = 0; q < 4; ++q)
        nxv[q] = xg[(long)b * G_DIM + hid + 8 * q];
    float nkp = 1.0f - (float)reset[b];

    for (int t = 0; t < T_DIM; ++t) {
        float xv[4];
        #pragma unroll
        for (int q = 0; q < 4; ++q) xv[q] = nxv[q];
        const float kp = nkp;

        // Prefetch t+1 (hits L2)
        if (t + 1 < T_DIM) {
            const long base = ((long)(t + 1) * B_DIM + b) * G_DIM;
            #pragma unroll
            for (int q = 0; q < 4; ++q)
                nxv[q] = xg[base + hid + 8 * q];
            nkp = 1.0f - (float)reset[(t + 1) * B_DIM + b];
        }

        // Apply reset mask, then gates = xg + h@Whh^T + bhh
        const float hk = h * kp;
        c *= kp;

        float g0 = xv[0] + bias[0];
        float g1 = xv[1] + bias[1];
        float g2 = xv[2] + bias[2];
        float g3 = xv[3] + bias[3];
        #pragma unroll
        for (int j = 0; j < 8; ++j) {
            const float hj = __shfl(hk, lb + j, 32);
            g0 = fmaf(hj, w[0][j], g0);
            g1 = fmaf(hj, w[1][j], g1);
            g2 = fmaf(hj, w[2][j], g2);
            g3 = fmaf(hj, w[3][j], g3);
        }

        const float ig = fast_sigmoid(g0);
        const float fg = fast_sigmoid(g1);
        const float gg = fast_tanh(g2);
        const float og = fast_sigmoid(g3);
        c = fmaf(fg, c, ig * gg);
        h = og * fast_tanh(c);

        // value[t,b] = dot(h, Wproj) + bproj via 8-lane butterfly
        float v = h * wp;
        v += __shfl_xor(v, 1, 32);
        v += __shfl_xor(v, 2, 32);
        v += __shfl_xor(v, 4, 32);
        if (hid == 0)
            out[(long)t * B_DIM + b] = v + bp;
    }
}

// ---------------------------------------------------------------------------
extern "C" void kernel_launch(void* const* d_in, const int* in_sizes, int n_in,
                              void* d_out, int out_size, void* d_ws, size_t ws_size,
                              hipStream_t stream) {
    const float* x     = (const float*)d_in[0];
    const int*   reset = (const int*)  d_in[1];
    const float* Wih   = (const float*)d_in[2];
    const float* Whh   = (const float*)d_in[3];
    const float* bih   = (const float*)d_in[4];
    const float* bhh   = (const float*)d_in[5];
    const float* Wproj = (const float*)d_in[6];
    const float* bproj = (const float*)d_in[7];
    float* out = (float*)d_out;

    float* xgates = (float*)d_ws;  // T*B*32 f32 = 64 MiB scratch

    // Phase 1: 524288 rows / 128 rows per block = 4096 blocks
    xgates_gemm_kernel<<<(T_DIM * B_DIM) / 128, 256, 0, stream>>>(x, Wih, bih, xgates);
    // Phase 2: 256 batches / (4 per wave * 8 waves) = 8 blocks
    lstm_scan_kernel<<<B_DIM / 32, 256, 0, stream>>>(xgates, reset, Whh, bhh,
                                                     Wproj, bproj, out);
}